// IMLESampler_15504831938714
// MI455X (gfx1250) — compile-verified
//
#include <hip/hip_runtime.h>
#include <stdint.h>

// Problem geometry (fixed by the reference): S=16, B=128, N=16384, k=32.
#define N_ELEM 16384
#define BLOCK  512
#define IPT    (N_ELEM / BLOCK)   // 32 elements per thread
#define NVEC   (IPT / 4)          // 8 x float4 per thread

typedef __attribute__((ext_vector_type(4))) float        fvec4;
typedef __attribute__((ext_vector_type(4))) unsigned int uvec4;

// Monotone map float -> uint32 so that (a > b) <=> (key(a) > key(b)).
__device__ __forceinline__ uint32_t f2key(float f) {
  uint32_t u = __float_as_uint(f);
  return (u & 0x80000000u) ? ~u : (u | 0x80000000u);
}

__device__ __forceinline__ uint32_t perturb_key(float u, float logit) {
  // gumbel = -log(-log(u)); perturbed = logit + gumbel
  float p = logit - __logf(-__logf(u));
  return f2key(p);
}

__launch_bounds__(BLOCK)
__global__ void imle_topk_kernel(const float* __restrict__ logits,
                                 const float* __restrict__ noise,
                                 const int*   __restrict__ kptr,
                                 float* __restrict__ out,
                                 int B)
{
  __shared__ __align__(16) uint32_t keys[N_ELEM];   // 64 KB: one full row of keys
  __shared__ uint32_t hist[256];
  __shared__ uint32_t ssum[256];
  __shared__ uint32_t sh_prefix;
  __shared__ uint32_t sh_krem;
  __shared__ uint32_t sh_tie;

  const int row = blockIdx.x;        // s*B + b
  // B is 128 (power of two) in this harness; keep a safe fallback.
  const int b   = (B & (B - 1)) ? (row % B) : (row & (B - 1));
  const int t   = threadIdx.x;

  const float* __restrict__ nrow = noise  + (size_t)row * N_ELEM;
  const float* __restrict__ lrow = logits + (size_t)b   * N_ELEM;
  float*       __restrict__ orow = out    + (size_t)row * N_ELEM;

  // ---------------- Phase A: stage noise row into LDS via async DMA ----------
  // Each lane issues 8 x b128 async copies of ITS OWN elements. Non-temporal
  // hint: the 128 MiB noise stream is touched exactly once — keep it from
  // evicting the (16x reused) 8 MiB logits working set out of L2.
#pragma unroll
  for (int i = 0; i < NVEC; ++i) {
    const int e = (i * BLOCK + t) * 4;
    const uint64_t ga = (uint64_t)(uintptr_t)(nrow + e);
    const uint32_t la = (uint32_t)(uintptr_t)(&keys[e]);  // low 32 bits = LDS offset
    asm volatile("global_load_async_to_lds_b128 %0, %1, off th:TH_LOAD_NT"
                 :: "v"(la), "v"(ga) : "memory");
  }

  // Preload all logits chunks NOW so their global loads overlap the async DMA
  // (the s_wait_asynccnt asm below has a memory clobber and would otherwise
  // prevent hoisting them).
  fvec4 lg[NVEC];
#pragma unroll
  for (int i = 0; i < NVEC; ++i) {
    const int e = (i * BLOCK + t) * 4;
    lg[i] = *reinterpret_cast<const fvec4*>(lrow + e);
  }

  // Pipelined consume: async loads of one wave complete IN ORDER, so chunk i
  // is resident once ASYNCcnt <= NVEC-1-i. Transform chunk 0 while chunks
  // 1..7 are still in flight.
#pragma unroll
  for (int i = 0; i < NVEC; ++i) {
    asm volatile("s_wait_asynccnt %0" :: "i"(NVEC - 1 - i) : "memory");
    const int e = (i * BLOCK + t) * 4;
    uvec4 ub = *reinterpret_cast<const uvec4*>(&keys[e]);       // ds_load_b128
    uvec4 kk;
    kk.x = perturb_key(__uint_as_float(ub.x), lg[i].x);
    kk.y = perturb_key(__uint_as_float(ub.y), lg[i].y);
    kk.z = perturb_key(__uint_as_float(ub.z), lg[i].z);
    kk.w = perturb_key(__uint_as_float(ub.w), lg[i].w);
    *reinterpret_cast<uvec4*>(&keys[e]) = kk;                   // ds_store_b128
  }

  const uint32_t k = (uint32_t)*kptr;

  // ---------------- Phase B: radix-select the k-th largest key ---------------
  uint32_t prefix = 0u;     // known high bits of threshold key
  uint32_t pmask  = 0u;     // mask of known bits
  uint32_t krem   = k;      // rank remaining within current prefix class

  for (int pass = 0; pass < 4; ++pass) {
    const int shift = 24 - 8 * pass;

    if (t < 256) hist[t] = 0u;
    __syncthreads();                      // also fences Phase A key writes

#pragma unroll 4
    for (int i = 0; i < IPT; ++i) {
      uint32_t key = keys[i * BLOCK + t];
      if ((key & pmask) == prefix)
        atomicAdd(&hist[(key >> shift) & 0xFFu], 1u);   // ds_add_u32
    }
    __syncthreads();

    // Inclusive suffix sum: ssum[d] = #keys (in class) with digit >= d.
    if (t < 256) ssum[t] = hist[t];
    __syncthreads();
    for (int off = 1; off < 256; off <<= 1) {
      uint32_t v = 0u;
      if (t < 256 && (t + off) < 256) v = ssum[t + off];
      __syncthreads();
      if (t < 256) ssum[t] += v;
      __syncthreads();
    }

    if (t < 256) {
      const uint32_t incl = ssum[t];          // digit >= t
      const uint32_t gt   = incl - hist[t];   // digit >  t
      if (gt < krem && incl >= krem) {        // exactly one digit satisfies
        sh_prefix = prefix | ((uint32_t)t << shift);
        sh_krem   = krem - gt;
      }
    }
    __syncthreads();
    prefix = sh_prefix;
    krem   = sh_krem;
    pmask |= (0xFFu << shift);
  }

  // prefix == exact key of the k-th largest element; krem == how many
  // elements EQUAL to the threshold belong in the top-k.
  const uint32_t T     = prefix;
  const uint32_t nties = krem;

  if (t == 0) sh_tie = 0u;
  __syncthreads();

  // ---------------- Phase C: emit hard mask (non-temporal b128 stores) -------
#pragma unroll
  for (int i = 0; i < NVEC; ++i) {
    const int e = (i * BLOCK + t) * 4;
    uvec4 kk = *reinterpret_cast<const uvec4*>(&keys[e]);
    fvec4 m;
    m.x = (kk.x > T) ? 1.0f : 0.0f;
    m.y = (kk.y > T) ? 1.0f : 0.0f;
    m.z = (kk.z > T) ? 1.0f : 0.0f;
    m.w = (kk.w > T) ? 1.0f : 0.0f;
    if (kk.x == T) m.x = (atomicAdd(&sh_tie, 1u) < nties) ? 1.0f : 0.0f;
    if (kk.y == T) m.y = (atomicAdd(&sh_tie, 1u) < nties) ? 1.0f : 0.0f;
    if (kk.z == T) m.z = (atomicAdd(&sh_tie, 1u) < nties) ? 1.0f : 0.0f;
    if (kk.w == T) m.w = (atomicAdd(&sh_tie, 1u) < nties) ? 1.0f : 0.0f;
    __builtin_nontemporal_store(m, reinterpret_cast<fvec4*>(orow + e));
  }
}

extern "C" void kernel_launch(void* const* d_in, const int* in_sizes, int n_in,
                              void* d_out, int out_size, void* d_ws, size_t ws_size,
                              hipStream_t stream) {
  const float* logits = (const float*)d_in[0];   // [B, N] f32
  const float* noise  = (const float*)d_in[1];   // [S, B, N] f32
  const int*   kptr   = (const int*)d_in[2];     // scalar k (device)
  float*       out    = (float*)d_out;           // [S, B, N] f32

  const int B    = in_sizes[0] / N_ELEM;         // 128
  const int rows = in_sizes[1] / N_ELEM;         // S*B = 2048

  imle_topk_kernel<<<dim3(rows), dim3(BLOCK), 0, stream>>>(
      logits, noise, kptr, out, B);
}